// SampleRepresentativeCalculator_18820546691128
// MI455X (gfx1250) — compile-verified
//
#include <hip/hip_runtime.h>
#include <cstdint>
#include <cstddef>

// ---------------------------------------------------------------------------
// SampleRepresentativeCalculator — elementwise, HBM-bandwidth-bound.
//   traffic: 20 B/elem * 29.36M elem ~= 0.587 GB  ->  ~25 us floor @ 23.3 TB/s
// Strategy (CDNA5/gfx1250):
//   * b128 per-lane transfers (4 x f32 / 4 x i32 per lane)
//   * double-buffered GLOBAL_LOAD_ASYNC_TO_LDS_B128 (th:TH_LOAD_NT) prefetch
//     pipeline, per-wave synchronized with s_wait_asynccnt (no workgroup
//     barriers: each lane reads back exactly the 16B it fetched)
//   * consumption via true ds_load_b128 (typed addrspace(3) arrays)
//   * non-temporal b128 stores for the write-once outputs (footprint > L2)
//   * no redundant tail prefetch: uniform branch + s_wait_asynccnt 0 on the
//     final iteration (saves ~84 MB of re-read traffic vs. clamping)
// ---------------------------------------------------------------------------

typedef float v4f __attribute__((ext_vector_type(4)));
typedef int   v4i __attribute__((ext_vector_type(4)));

#define TPB   256
#define VEC   4
#define TILE  (TPB * VEC)           // 1024 elements per tile

struct alignas(16) Smem {
    v4f o[2][TPB];   // originals, double buffered
    v4f p[2][TPB];   // predictions
    v4i m[2][TPB];   // max_errors
};

__device__ __forceinline__ void async_ld_b128(uint32_t lds_off, const void* gaddr) {
    // GLOBAL_LOAD_ASYNC_TO_LDS_B128: VDST = per-lane LDS byte offset,
    // VADDR = per-lane 64-bit global address, saddr = off (GV mode).
    // Tracked by ASYNCcnt (CDNA5 ISA). NT: read-once
    // stream (352 MB) larger than the 192 MB L2.
    asm volatile("global_load_async_to_lds_b128 %0, %1, off th:TH_LOAD_NT"
                 :
                 : "v"(lds_off), "v"((uint64_t)(uintptr_t)gaddr)
                 : "memory");
}

__global__ void __launch_bounds__(TPB) srep_main(
    const float* __restrict__ orig, const float* __restrict__ pred,
    const int*   __restrict__ merr, const float* __restrict__ phi,
    const float* __restrict__ psi,  const float* __restrict__ theta,
    float* __restrict__ reps_out,   float* __restrict__ bc_out,
    unsigned numTiles, unsigned xy)
{
    __shared__ Smem sm;   // 24 KB

    const unsigned tid = threadIdx.x;
    const float    th  = theta[0];
    const float    thE = th + 1e-8f;

    unsigned t = blockIdx.x;
    if (t >= numTiles) return;
    const unsigned stride = gridDim.x;

    // Per-lane LDS byte offsets of this lane's slots (generic-pointer low
    // 32 bits == offset within the workgroup LDS allocation, which is what
    // the async load's VDST expects; hardware adds LDS_BASE).
    const uint32_t lo0 = (uint32_t)(uintptr_t)&sm.o[0][tid];
    const uint32_t lo1 = (uint32_t)(uintptr_t)&sm.o[1][tid];
    const uint32_t lp0 = (uint32_t)(uintptr_t)&sm.p[0][tid];
    const uint32_t lp1 = (uint32_t)(uintptr_t)&sm.p[1][tid];
    const uint32_t lm0 = (uint32_t)(uintptr_t)&sm.m[0][tid];
    const uint32_t lm1 = (uint32_t)(uintptr_t)&sm.m[1][tid];

    // Prime the pipeline: tile t -> stage 0.  ASYNCcnt = 3 after this.
    {
        const unsigned e = t * (unsigned)TILE + tid * VEC;
        async_ld_b128(lo0, orig + e);
        async_ld_b128(lp0, pred + e);
        async_ld_b128(lm0, merr + e);
    }

    unsigned stage = 0u;
    for (; t < numTiles; t += stride) {
        // Prefetch the next tile into the other stage. t/stride/numTiles are
        // wave-uniform, so this branch keeps EXEC all-ones for the async ops.
        const unsigned nt = t + stride;
        if (nt < numTiles) {
            const unsigned e = nt * (unsigned)TILE + tid * VEC;
            // The target buffer was last *read* (ds_load) an iteration ago;
            // drain DScnt before the async engine may overwrite it.
            asm volatile("s_wait_dscnt 0" ::: "memory");
            if (stage == 0u) {
                async_ld_b128(lo1, orig + e);
                async_ld_b128(lp1, pred + e);
                async_ld_b128(lm1, merr + e);
            } else {
                async_ld_b128(lo0, orig + e);
                async_ld_b128(lp0, pred + e);
                async_ld_b128(lm0, merr + e);
            }
            // 6 outstanding; async loads complete in order, so <=3 means the
            // 3 loads for the *current* tile have landed in LDS.
            asm volatile("s_wait_asynccnt 3" ::: "memory");
        } else {
            // Last iteration: only this tile's 3 loads are outstanding.
            asm volatile("s_wait_asynccnt 0" ::: "memory");
        }

        // True LDS consumption: ds_load_b128 from addrspace(3) arrays.
        const v4f o = sm.o[stage][tid];
        const v4f p = sm.p[stage][tid];
        const v4i m = sm.m[stage][tid];

        const unsigned e = t * (unsigned)TILE + tid * VEC;  // N < 2^32
        const unsigned z = e / xy;            // 4 consecutive elems share z
        const float ph = phi[z];
        const float ps = psi[z];
        const bool  zero_adj = (ph == 0.0f) && (ps == 0.0f);

        v4f reps, bc;
#pragma unroll
        for (int j = 0; j < 4; ++j) {
            const float oj = o[j];
            const float pj = p[j];
            const int   mj = m[j];
            const float step = 2.0f * (float)mj + 1.0f;
            // jnp.round == round-half-to-even == rintf under default RNE mode
            const float qres = __builtin_rintf((oj - pj) / step) * step;
            const float bcj  = (mj == 0) ? oj : (pj + qres);
            const float d    = pj - bcj;
            const float ad   = __builtin_fabsf(d);
            // jnp.sign: sign(0) == 0
            const float sgn  = (d > 0.0f) ? 1.0f : ((d < 0.0f) ? -1.0f : 0.0f);
            const float adj  = (ad <= th) ? (ph * d / thE)
                                          : (ps * sgn * (ad - th) / (ad + 1e-8f));
            bc[j]   = bcj;
            reps[j] = zero_adj ? bcj : (bcj + adj);
        }
        // Streaming outputs: footprint >> 192 MB L2 -> non-temporal b128.
        __builtin_nontemporal_store(reps, (v4f*)(reps_out + e));
        __builtin_nontemporal_store(bc,   (v4f*)(bc_out  + e));

        stage ^= 1u;
    }
}

// Scalar tail (not taken for the reference shape: N % 1024 == 0), kept for
// robustness.
__global__ void srep_tail(
    const float* __restrict__ orig, const float* __restrict__ pred,
    const int*   __restrict__ merr, const float* __restrict__ phi,
    const float* __restrict__ psi,  const float* __restrict__ theta,
    float* __restrict__ reps_out,   float* __restrict__ bc_out,
    unsigned start, unsigned N, unsigned xy)
{
    const unsigned i = start + blockIdx.x * blockDim.x + threadIdx.x;
    if (i >= N) return;
    const float th  = theta[0];
    const float thE = th + 1e-8f;
    const float oj = orig[i];
    const float pj = pred[i];
    const int   mj = merr[i];
    const float step = 2.0f * (float)mj + 1.0f;
    const float qres = __builtin_rintf((oj - pj) / step) * step;
    const float bcj  = (mj == 0) ? oj : (pj + qres);
    const float d    = pj - bcj;
    const float ad   = __builtin_fabsf(d);
    const float sgn  = (d > 0.0f) ? 1.0f : ((d < 0.0f) ? -1.0f : 0.0f);
    const unsigned z = i / xy;
    const float ph = phi[z];
    const float ps = psi[z];
    const float adj = (ad <= th) ? (ph * d / thE)
                                 : (ps * sgn * (ad - th) / (ad + 1e-8f));
    const float rj = ((ph == 0.0f) && (ps == 0.0f)) ? bcj : (bcj + adj);
    reps_out[i] = rj;
    bc_out[i]   = bcj;
}

extern "C" void kernel_launch(void* const* d_in, const int* in_sizes, int n_in,
                              void* d_out, int out_size, void* d_ws, size_t ws_size,
                              hipStream_t stream)
{
    (void)n_in; (void)out_size; (void)d_ws; (void)ws_size;

    const float* orig  = (const float*)d_in[0];
    const float* pred  = (const float*)d_in[1];
    const int*   merr  = (const int*)  d_in[2];
    const float* phi   = (const float*)d_in[3];
    const float* psi   = (const float*)d_in[4];
    const float* theta = (const float*)d_in[5];

    const unsigned N    = (unsigned)in_sizes[0];   // 29,360,128
    const unsigned Zdim = (unsigned)in_sizes[3];   // 224
    const unsigned xy   = N / Zdim;                // 131072 (= Y*X)

    float* reps_out = (float*)d_out;
    float* bc_out   = (float*)d_out + N;

    const unsigned numTiles = N / TILE;            // 28,672
    const unsigned rem      = N % TILE;            // 0 for reference shape

    if (numTiles) {
        unsigned grid = numTiles < 7168u ? numTiles : 7168u;  // 4 tiles/block
        hipLaunchKernelGGL(srep_main, dim3(grid), dim3(TPB), 0, stream,
                           orig, pred, merr, phi, psi, theta,
                           reps_out, bc_out, numTiles, xy);
    }
    if (rem) {
        const unsigned start  = numTiles * TILE;
        const unsigned blocks = (rem + TPB - 1u) / TPB;
        hipLaunchKernelGGL(srep_tail, dim3(blocks), dim3(TPB), 0, stream,
                           orig, pred, merr, phi, psi, theta,
                           reps_out, bc_out, start, N, xy);
    }
}